// RGNN_72799695667753
// MI455X (gfx1250) — compile-verified
//
#include <hip/hip_runtime.h>

// ---------------------------------------------------------------------------
// Hetero 2-layer GAT (paper/author, 3 relations), fp32 end-to-end.
// GEMMs use CDNA5 V_WMMA_F32_16X16X4_F32 (full fp32 matrix path).
// Edge softmax/aggregation uses native float global atomics.
// ---------------------------------------------------------------------------

#define NP 200000
#define NA 100000
#define HID 128
#define NHEAD 4
#define CH 32

typedef float v2f __attribute__((ext_vector_type(2)));
typedef float v8f __attribute__((ext_vector_type(8)));

// ---------------------------------------------------------------------------
// fp32 WMMA GEMM: C[M,N] = A[M,K] @ B[K,N] (+ bias[N] if bias != nullptr)
// Row-major everything. Requires M%16==0, K%4==0, N==NT*16.
// Block = 128 threads = 4 waves; each wave owns a 16-row strip and NT col-tiles.
// A-frag layout (16x4 f32): lanes 0-15 -> M=lane, K=k..k+1 ; lanes 16-31 -> K=k+2..k+3
// B-frag layout (4x16 f32): VGPR i: lanes 0-15 -> K=k+i ; lanes 16-31 -> K=k+2+i
// C/D layout (16x16 f32): VGPR i: lanes 0-15 -> M=i ; lanes 16-31 -> M=8+i ; N=lane&15
// ---------------------------------------------------------------------------
template <int NT>
__global__ __launch_bounds__(128)
void gemm_f32_wmma(const float* __restrict__ A, const float* __restrict__ B,
                   const float* __restrict__ bias, float* __restrict__ C,
                   int M, int K, int N) {
  const int wave = threadIdx.x >> 5;
  const int lane = threadIdx.x & 31;
  const int row0 = (blockIdx.x * 4 + wave) * 16;
  if (row0 >= M) return;                   // wave-uniform exit; EXEC all-ones below
  const int half = lane >> 4;              // 0: K lo pair, 1: K hi pair
  const int l15  = lane & 15;

  v8f acc[NT];
#pragma unroll
  for (int n = 0; n < NT; ++n) acc[n] = (v8f){0.f,0.f,0.f,0.f,0.f,0.f,0.f,0.f};

  const float* arow = A + (size_t)(row0 + l15) * K;

  for (int k = 0; k < K; k += 4) {
    const int ka = k + half * 2;
    const float2 av = *(const float2*)(arow + ka);   // 8B aligned (ka even)
    v2f afrag; afrag.x = av.x; afrag.y = av.y;
#pragma unroll
    for (int n = 0; n < NT; ++n) {
      const int col = n * 16 + l15;
      v2f bfrag;
      bfrag.x = B[(size_t)(ka + 0) * N + col];
      bfrag.y = B[(size_t)(ka + 1) * N + col];
      acc[n] = __builtin_amdgcn_wmma_f32_16x16x4_f32(
          false, afrag, false, bfrag, (short)0, acc[n], false, false);
    }
  }

#pragma unroll
  for (int n = 0; n < NT; ++n) {
    const int col = n * 16 + l15;
    const float bv = bias ? bias[col] : 0.0f;
#pragma unroll
    for (int i = 0; i < 8; ++i) {
      const int row = row0 + half * 8 + i;
      C[(size_t)row * N + col] = acc[n][i] + bv;
    }
  }
}

// ---------------------------------------------------------------------------
// a[n,h] = sum_c H[n, h*CH + c] * att[h*CH + c]
// ---------------------------------------------------------------------------
__global__ void att_scores(const float* __restrict__ Hm, const float* __restrict__ att,
                           float* __restrict__ out, int Nn) {
  const long long idx = (long long)blockIdx.x * blockDim.x + threadIdx.x;
  if (idx >= (long long)Nn * NHEAD) return;
  const int n = (int)(idx >> 2);
  const int h = (int)(idx & 3);
  const float4* hp = (const float4*)(Hm + (size_t)n * HID + h * CH);
  const float4* ap = (const float4*)(att + h * CH);
  float s = 0.f;
#pragma unroll
  for (int i = 0; i < CH / 4; ++i) {
    const float4 a = hp[i], b = ap[i];
    s += a.x * b.x + a.y * b.y + a.z * b.z + a.w * b.w;
  }
  out[idx] = s;
}

// ---------------------------------------------------------------------------
// fills
// ---------------------------------------------------------------------------
__global__ void fill_val(float* __restrict__ p, float v, long long n) {
  const long long i = (long long)blockIdx.x * blockDim.x + threadIdx.x;
  if (i < n) p[i] = v;
}

__global__ void fill_bias2(float* __restrict__ out, const float* __restrict__ b1,
                           const float* __restrict__ b2, long long total) {
  const long long i = (long long)blockIdx.x * blockDim.x + threadIdx.x;
  if (i >= total) return;
  const int j = (int)(i & (HID - 1));
  float v = b1[j];
  if (b2) v += b2[j];
  out[i] = v;
}

__global__ void leaky_inplace(float* __restrict__ p, float slope, long long n) {
  const long long i = (long long)blockIdx.x * blockDim.x + threadIdx.x;
  if (i >= n) return;
  const float v = p[i];
  p[i] = v > 0.f ? v : slope * v;
}

__global__ void fix_m(float* __restrict__ m, long long n) {
  const long long i = (long long)blockIdx.x * blockDim.x + threadIdx.x;
  if (i >= n) return;
  const float v = m[i];
  if (v < -3.0e38f) m[i] = 0.f;   // -inf (no in-edges) -> 0, matches reference
}

// ---------------------------------------------------------------------------
// edge pass 1: e = leaky_relu(a_s[src]+a_d[dst], 0.2); segment max via atomicMax
// ---------------------------------------------------------------------------
__global__ void edge_logits(const int* __restrict__ src, const int* __restrict__ dst,
                            const float* __restrict__ as_, const float* __restrict__ ad_,
                            float* __restrict__ eb, float* __restrict__ mb, int E) {
  const int e = blockIdx.x * blockDim.x + threadIdx.x;
  if (e >= E) return;
  const int s = src[e], d = dst[e];
  const float4 av = *(const float4*)(as_ + (size_t)s * NHEAD);
  const float4 dv = *(const float4*)(ad_ + (size_t)d * NHEAD);
  float4 ev;
  ev.x = av.x + dv.x; ev.x = ev.x > 0.f ? ev.x : 0.2f * ev.x;
  ev.y = av.y + dv.y; ev.y = ev.y > 0.f ? ev.y : 0.2f * ev.y;
  ev.z = av.z + dv.z; ev.z = ev.z > 0.f ? ev.z : 0.2f * ev.z;
  ev.w = av.w + dv.w; ev.w = ev.w > 0.f ? ev.w : 0.2f * ev.w;
  *(float4*)(eb + (size_t)e * NHEAD) = ev;
  float* mrow = mb + (size_t)d * NHEAD;
  atomicMax(mrow + 0, ev.x);
  atomicMax(mrow + 1, ev.y);
  atomicMax(mrow + 2, ev.z);
  atomicMax(mrow + 3, ev.w);
}

// ---------------------------------------------------------------------------
// edge pass 2: ex = exp(e - m[dst]); eb <- ex; segment sum via atomicAdd
// ---------------------------------------------------------------------------
__global__ void edge_exp(const int* __restrict__ dst, const float* __restrict__ mb,
                         float* __restrict__ eb, float* __restrict__ sb, int E) {
  const int e = blockIdx.x * blockDim.x + threadIdx.x;
  if (e >= E) return;
  const int d = dst[e];
  const float4 ev = *(const float4*)(eb + (size_t)e * NHEAD);
  const float4 mv = *(const float4*)(mb + (size_t)d * NHEAD);
  float4 ex;
  ex.x = __expf(ev.x - mv.x);
  ex.y = __expf(ev.y - mv.y);
  ex.z = __expf(ev.z - mv.z);
  ex.w = __expf(ev.w - mv.w);
  *(float4*)(eb + (size_t)e * NHEAD) = ex;
  float* srow = sb + (size_t)d * NHEAD;
  atomicAdd(srow + 0, ex.x);
  atomicAdd(srow + 1, ex.y);
  atomicAdd(srow + 2, ex.z);
  atomicAdd(srow + 3, ex.w);
}

// ---------------------------------------------------------------------------
// edge pass 3: out[dst] += hs[src] * alpha  (one wave32 per edge, float4/lane)
// ---------------------------------------------------------------------------
__global__ void edge_aggr(const int* __restrict__ src, const int* __restrict__ dst,
                          const float* __restrict__ eb, const float* __restrict__ sb,
                          const float* __restrict__ HS, float* __restrict__ OUT, int E) {
  const long long t = (long long)blockIdx.x * blockDim.x + threadIdx.x;
  const long long e = t >> 5;
  if (e >= E) return;
  const int lane = (int)(t & 31);
  const int s = src[e], d = dst[e];
  const int h = lane >> 3;  // 4 channels/lane -> head = (lane*4)/32
  const float alpha = eb[e * NHEAD + h] / (sb[(size_t)d * NHEAD + h] + 1e-16f);
  const float4 hv = *(const float4*)(HS + (size_t)s * HID + lane * 4);
  float* o = OUT + (size_t)d * HID + lane * 4;
  atomicAdd(o + 0, hv.x * alpha);
  atomicAdd(o + 1, hv.y * alpha);
  atomicAdd(o + 2, hv.z * alpha);
  atomicAdd(o + 3, hv.w * alpha);
}

// ---------------------------------------------------------------------------
// host-side helpers
// ---------------------------------------------------------------------------
static inline void gemm(const float* A, const float* B, const float* bias, float* C,
                        int M, int K, int N, hipStream_t st) {
  const int blocks = (M / 16 + 3) / 4;
  if (N == 128)      gemm_f32_wmma<8><<<blocks, 128, 0, st>>>(A, B, bias, C, M, K, N);
  else /* N == 64 */ gemm_f32_wmma<4><<<blocks, 128, 0, st>>>(A, B, bias, C, M, K, N);
}

static inline int gdim(long long n, int b) { return (int)((n + b - 1) / b); }

static void run_relation(const float* HSp, const float* HDp, int Ns, int Nd,
                         const int* src, const int* dst, int E,
                         const float* att_s, const float* att_d,
                         float* ASb, float* ADb, float* Mb, float* Sb, float* EB,
                         float* OUT, hipStream_t st) {
  att_scores<<<gdim((long long)Ns * NHEAD, 256), 256, 0, st>>>(HSp, att_s, ASb, Ns);
  att_scores<<<gdim((long long)Nd * NHEAD, 256), 256, 0, st>>>(HDp, att_d, ADb, Nd);
  fill_val<<<gdim((long long)Nd * NHEAD, 256), 256, 0, st>>>(Mb, -3.4e38f, (long long)Nd * NHEAD);
  fill_val<<<gdim((long long)Nd * NHEAD, 256), 256, 0, st>>>(Sb, 0.f, (long long)Nd * NHEAD);
  edge_logits<<<gdim(E, 256), 256, 0, st>>>(src, dst, ASb, ADb, EB, Mb, E);
  fix_m<<<gdim((long long)Nd * NHEAD, 256), 256, 0, st>>>(Mb, (long long)Nd * NHEAD);
  edge_exp<<<gdim(E, 256), 256, 0, st>>>(dst, Mb, EB, Sb, E);
  edge_aggr<<<gdim((long long)E * 32, 256), 256, 0, st>>>(src, dst, EB, Sb, HSp, OUT, E);
}

static void hetero_layer(const float* XPin, const float* XAin, float* XPout, float* XAout,
                         const float* W, const float* As, const float* Ad, const float* B,
                         const int* sc, const int* dc, int Ec,
                         const int* sw, const int* dw, int Ew,
                         const int* swb, const int* dwb, int Ewb,
                         float* HS, float* HD, float* ASb, float* ADb,
                         float* Mb, float* Sb, float* EB, hipStream_t st) {
  // HeteroConv 'sum': paper accumulator starts at b[cites]+b[writes], author at b[wb]
  fill_bias2<<<gdim((long long)NP * HID, 256), 256, 0, st>>>(XPout, B + 0 * HID, B + 1 * HID,
                                                             (long long)NP * HID);
  fill_bias2<<<gdim((long long)NA * HID, 256), 256, 0, st>>>(XAout, B + 2 * HID, nullptr,
                                                             (long long)NA * HID);
  // relation 0: cites (paper -> paper); hs == hd
  gemm(XPin, W + 0 * HID * HID, nullptr, HS, NP, HID, HID, st);
  run_relation(HS, HS, NP, NP, sc, dc, Ec, As + 0 * HID, Ad + 0 * HID,
               ASb, ADb, Mb, Sb, EB, XPout, st);
  // relation 1: writes (author -> paper)
  gemm(XAin, W + 1 * HID * HID, nullptr, HS, NA, HID, HID, st);
  gemm(XPin, W + 1 * HID * HID, nullptr, HD, NP, HID, HID, st);
  run_relation(HS, HD, NA, NP, sw, dw, Ew, As + 1 * HID, Ad + 1 * HID,
               ASb, ADb, Mb, Sb, EB, XPout, st);
  // relation 2: written_by (paper -> author)
  gemm(XPin, W + 2 * HID * HID, nullptr, HS, NP, HID, HID, st);
  gemm(XAin, W + 2 * HID * HID, nullptr, HD, NA, HID, HID, st);
  run_relation(HS, HD, NP, NA, swb, dwb, Ewb, As + 2 * HID, Ad + 2 * HID,
               ASb, ADb, Mb, Sb, EB, XAout, st);
}

extern "C" void kernel_launch(void* const* d_in, const int* in_sizes, int n_in,
                              void* d_out, int out_size, void* d_ws, size_t ws_size,
                              hipStream_t stream) {
  const float* x_paper  = (const float*)d_in[0];
  const float* x_author = (const float*)d_in[1];
  const float* Wp   = (const float*)d_in[2];
  const float* bp   = (const float*)d_in[3];
  const float* Wa   = (const float*)d_in[4];
  const float* ba   = (const float*)d_in[5];
  const float* W0   = (const float*)d_in[6];
  const float* as0  = (const float*)d_in[7];
  const float* ad0  = (const float*)d_in[8];
  const float* b0   = (const float*)d_in[9];
  const float* W1   = (const float*)d_in[10];
  const float* as1  = (const float*)d_in[11];
  const float* ad1  = (const float*)d_in[12];
  const float* b1   = (const float*)d_in[13];
  const float* Wlin = (const float*)d_in[14];
  const float* blin = (const float*)d_in[15];
  const int* sc  = (const int*)d_in[16];
  const int* dc  = (const int*)d_in[17];
  const int* sw  = (const int*)d_in[18];
  const int* dw  = (const int*)d_in[19];
  const int* swb = (const int*)d_in[20];
  const int* dwb = (const int*)d_in[21];
  const int Ec  = in_sizes[16];
  const int Ew  = in_sizes[18];
  const int Ewb = in_sizes[20];

  float* ws = (float*)d_ws;
  size_t off = 0;
  auto alloc = [&](size_t nflt) { float* p = ws + off; off += nflt; return p; };
  float* XP_A = alloc((size_t)NP * HID);
  float* XP_B = alloc((size_t)NP * HID);
  float* XA_A = alloc((size_t)NA * HID);
  float* XA_B = alloc((size_t)NA * HID);
  float* HS   = alloc((size_t)NP * HID);
  float* HD   = alloc((size_t)NP * HID);
  float* ASb  = alloc((size_t)NP * NHEAD);
  float* ADb  = alloc((size_t)NP * NHEAD);
  float* Mb   = alloc((size_t)NP * NHEAD);
  float* Sb   = alloc((size_t)NP * NHEAD);
  float* EB   = alloc((size_t)1000000 * NHEAD);

  // input projections
  gemm(x_paper,  Wp, bp, XP_A, NP, 256, HID, stream);
  gemm(x_author, Wa, ba, XA_A, NA, HID, HID, stream);

  // layer 0: A -> B
  hetero_layer(XP_A, XA_A, XP_B, XA_B, W0, as0, ad0, b0,
               sc, dc, Ec, sw, dw, Ew, swb, dwb, Ewb,
               HS, HD, ASb, ADb, Mb, Sb, EB, stream);
  leaky_inplace<<<gdim((long long)NP * HID, 256), 256, 0, stream>>>(XP_B, 0.01f, (long long)NP * HID);
  leaky_inplace<<<gdim((long long)NA * HID, 256), 256, 0, stream>>>(XA_B, 0.01f, (long long)NA * HID);

  // layer 1: B -> A
  hetero_layer(XP_B, XA_B, XP_A, XA_A, W1, as1, ad1, b1,
               sc, dc, Ec, sw, dw, Ew, swb, dwb, Ewb,
               HS, HD, ASb, ADb, Mb, Sb, EB, stream);

  // classifier on paper nodes
  gemm(XP_A, Wlin, blin, (float*)d_out, NP, HID, 64, stream);
}